// ChoquetIntegralConstrained_31782757991068
// MI455X (gfx1250) — compile-verified
//
#include <hip/hip_runtime.h>
#include <math.h>
#include <cstdint>

// ---------------------------------------------------------------------------
// Choquet integral (constrained) — MI455X / gfx1250
//   out[b] = sigmoid( (x[b,:] . w_eff) / sum(w_eff) - thr )
// Memory-bound GEMV: 541 MB of x streamed once, ~23 us floor @ 23.3 TB/s.
// V_WMMA_F32_16X16X4_F32 is the MAC engine (16-row tiles, B columns all equal
// to the same w chunk). x is staged via global_load_async_to_lds_b128 into a
// 4-stage LDS pipeline tracked with ASYNCcnt, keeping 48 KB of HBM traffic in
// flight per workgroup without burning VGPRs.
// ---------------------------------------------------------------------------

#define N_CRIT   128
#define N_PAIRS  8128
#define K_TOT    (N_CRIT + N_PAIRS)      // 8256
#define BATCH    16384
#define MIN_W    1e-7f

#define CK           64                  // K-chunk per pipeline stage
#define ROWS_PER_WG  64
#define WAVES_PER_WG 4                   // wave32: 4 waves * 16 rows
#define BLOCK        (WAVES_PER_WG * 32) // 128 threads
#define XSTRIDE      (CK + 4)            // 68 dwords: 16B-aligned rows, no LDS bank conflicts
#define NCHUNK       (K_TOT / CK)        // 129 (exact)
#define STAGES       4                   // async pipeline depth
#define APC          8                   // async instructions per chunk per wave

typedef __attribute__((ext_vector_type(2))) float v2f;
typedef __attribute__((ext_vector_type(8))) float v8f;

// s_wait_asynccnt with a compile-time immediate
#define WAIT_ASYNC(n) asm volatile("s_wait_asynccnt %0" :: "n"(n) : "memory")

// one wave-wide async 16B global->LDS copy (per-lane addresses)
__device__ __forceinline__ void async_b128(unsigned lds_off, const float* gaddr) {
    asm volatile("global_load_async_to_lds_b128 %0, %1, off"
                 :: "v"(lds_off), "v"(gaddr) : "memory");
}

// start offset of upper-triangle row i (k=1), N=128:  S(i) = i*(255-i)/2
__device__ __forceinline__ int tri_row_start(int i) {
    return (i * (2 * N_CRIT - 1 - i)) >> 1;
}

// ---------------------------------------------------------------------------
// Kernel 1: effective weights + 1/weight_sum + thr into workspace.
// ---------------------------------------------------------------------------
__global__ void __launch_bounds__(256)
choquet_prep_kernel(const float* __restrict__ wc,
                    const float* __restrict__ wint,
                    const float* __restrict__ thr,
                    float* __restrict__ w,        // [K_TOT]
                    float* __restrict__ scalars)  // [2]: 1/sum, thr
{
    __shared__ float red[256];
    const int tid = threadIdx.x;
    float local = 0.f;

    for (int i = tid; i < N_CRIT; i += 256) {
        float v = wc[i];
        v = (v < 0.f) ? MIN_W : v;
        w[i] = v;
        local += v;
    }

    for (int p = tid; p < N_PAIRS; p += 256) {
        float disc = 255.f * 255.f - 8.f * (float)p;
        int i = (int)((255.f - sqrtf(disc)) * 0.5f);
        if (i < 0) i = 0;
        if (i > N_CRIT - 2) i = N_CRIT - 2;
        while (i > 0 && tri_row_start(i) > p) --i;
        while (i < N_CRIT - 2 && tri_row_start(i + 1) <= p) ++i;
        const int j = i + 1 + (p - tri_row_start(i));

        float wi = wc[i]; wi = (wi < 0.f) ? MIN_W : wi;
        float wj = wc[j]; wj = (wj < 0.f) ? MIN_W : wj;
        float v = fmaxf(wint[p], fmaxf(-wi, -wj));
        w[N_CRIT + p] = v;
        local += v;
    }

    red[tid] = local;
    __syncthreads();
    for (int s = 128; s > 0; s >>= 1) {
        if (tid < s) red[tid] += red[tid + s];
        __syncthreads();
    }
    if (tid == 0) {
        scalars[0] = 1.f / red[0];
        scalars[1] = thr[0];
    }
}

// 16 WMMA MACs over one CK-wide chunk
__device__ __forceinline__ v8f wmma_chunk(v8f acc, const float* arow, const float* brow) {
#pragma unroll
    for (int kk = 0; kk < CK; kk += 4) {
        v2f a = *(const v2f*)(arow + kk);   // x[row][k..k+1] (lane-half dependent)
        v2f b = *(const v2f*)(brow + kk);   // w[k..k+1] broadcast to all columns
        acc = __builtin_amdgcn_wmma_f32_16x16x4_f32(
            /*neg_a=*/false, a, /*neg_b=*/false, b,
            /*c_mod=*/(short)0, acc, /*reuse_a=*/false, /*reuse_b=*/false);
    }
    return acc;
}

// ---------------------------------------------------------------------------
// Kernel 2: streaming GEMV, async-LDS pipelined, WMMA MACs.
// ---------------------------------------------------------------------------
__global__ void __launch_bounds__(BLOCK)
choquet_gemv_wmma_kernel(const float* __restrict__ x,
                         const float* __restrict__ w,
                         const float* __restrict__ scalars,
                         float* __restrict__ out)
{
    __shared__ float lds_w[K_TOT];                          // 33 KB weights
    __shared__ float lds_x[STAGES][ROWS_PER_WG * XSTRIDE];  // 4 x 17 KB x tiles

    const int tid     = threadIdx.x;
    const int lane    = tid & 31;
    const int wave    = tid >> 5;
    const int wg_row0 = blockIdx.x * ROWS_PER_WG;

    // cooperative load of w_eff into LDS (float4, coalesced)
    for (int i = tid; i < K_TOT / 4; i += BLOCK)
        ((float4*)lds_w)[i] = ((const float4*)w)[i];

    // staging assignment: 8 row-groups x 16 float4 columns per thread
    const int g  = tid >> 4;   // 0..7
    const int c4 = tid & 15;   // 0..15
    const float* gbase = x + (size_t)wg_row0 * K_TOT + c4 * 4;

    // per-thread LDS destinations (byte offsets) and per-buffer A-row bases
    const int m  = lane & 15;        // row within wave tile
    const int ks = (lane >> 4) * 2;  // K half select: 0 -> K0/K1, 1 -> K2/K3
    unsigned ldst[STAGES];
    const float* arow[STAGES];
#pragma unroll
    for (int s = 0; s < STAGES; ++s) {
        ldst[s] = (unsigned)(uintptr_t)&lds_x[s][g * XSTRIDE + c4 * 4];
        arow[s] = &lds_x[s][(wave * 16 + m) * XSTRIDE + ks];
    }

    // issue all APC async copies for one chunk into one stage buffer
    auto issue = [&](int chunk, int s) {
        const float* gp = gbase + (size_t)chunk * CK;
#pragma unroll
        for (int t = 0; t < APC; ++t)
            async_b128(ldst[s] + (unsigned)(t * 8 * XSTRIDE * 4),
                       gp + (size_t)(8 * t) * K_TOT);
    };

    // pipeline pre-fill: chunks 0..STAGES-1
#pragma unroll
    for (int s = 0; s < STAGES; ++s)
        issue(s, s);

    v8f acc = {};
    int c = 0;

    // steady state: outstanding = STAGES chunks; wait <= APC*(STAGES-1)
    // guarantees the oldest chunk landed (async loads complete in order).
    for (; c < NCHUNK - (STAGES - 1); ++c) {
        WAIT_ASYNC(APC * (STAGES - 1));
        __syncthreads();                       // all waves' portions visible
        acc = wmma_chunk(acc, arow[c & (STAGES - 1)], &lds_w[c * CK + ks]);
        __syncthreads();                       // all waves done reading buffer
        if (c + STAGES < NCHUNK)
            issue(c + STAGES, c & (STAGES - 1));
    }
    // tail: 3 remaining chunks, nothing left to issue, shrinking wait immediates
    WAIT_ASYNC(APC * 2);
    __syncthreads();
    acc = wmma_chunk(acc, arow[c & (STAGES - 1)], &lds_w[c * CK + ks]);
    __syncthreads();
    ++c;
    WAIT_ASYNC(APC * 1);
    __syncthreads();
    acc = wmma_chunk(acc, arow[c & (STAGES - 1)], &lds_w[c * CK + ks]);
    __syncthreads();
    ++c;
    WAIT_ASYNC(0);
    __syncthreads();
    acc = wmma_chunk(acc, arow[c & (STAGES - 1)], &lds_w[c * CK + ks]);

    // Epilogue: D columns identical; lanes 0 / 16 hold rows 0..7 / 8..15.
    const float inv = scalars[0];
    const float th  = scalars[1];
    if ((lane & 15) == 0) {
        const int rbase = wg_row0 + wave * 16 + (lane >> 4) * 8;
#pragma unroll
        for (int v = 0; v < 8; ++v) {
            float z = acc[v] * inv - th;
            out[rbase + v] = 1.f / (1.f + __expf(-z));
        }
    }
}

// ---------------------------------------------------------------------------
extern "C" void kernel_launch(void* const* d_in, const int* in_sizes, int n_in,
                              void* d_out, int out_size, void* d_ws, size_t ws_size,
                              hipStream_t stream)
{
    const float* x    = (const float*)d_in[0];  // [16384, 8256]
    const float* wc   = (const float*)d_in[1];  // [1, 128]
    const float* wint = (const float*)d_in[2];  // [1, 8128]
    const float* thr  = (const float*)d_in[3];  // [1]
    float* out = (float*)d_out;                 // [16384]

    float* w       = (float*)d_ws;   // K_TOT floats
    float* scalars = w + K_TOT;      // 2 floats

    choquet_prep_kernel<<<1, 256, 0, stream>>>(wc, wint, thr, w, scalars);
    choquet_gemv_wmma_kernel<<<BATCH / ROWS_PER_WG, BLOCK, 0, stream>>>(x, w, scalars, out);

    (void)in_sizes; (void)n_in; (void)out_size; (void)ws_size;
}